// MacroOverlap_71519795413200
// MI455X (gfx1250) — compile-verified
//
#include <hip/hip_runtime.h>
#include <hip/hip_bf16.h>

typedef float v2f __attribute__((ext_vector_type(2)));
typedef float v8f __attribute__((ext_vector_type(8)));

// piecewise-quadratic smooth overlap kernel; guarded so s<=0 (pad entries)
// contributes exactly 0 (avoids 0/0 NaN poisoning the sum).
__device__ __forceinline__ float bell_f(float d, float s) {
    if (!(d <= s) || s <= 0.0f) return 0.0f;
    float r   = d / s;
    float nr  = 1.0f - 2.0f * r * r;
    float fr  = 2.0f * (r - 1.0f) * (r - 1.0f);
    return (d <= 0.5f * s) ? nr : fr;
}

// ---------------------------------------------------------------------------
// Kernel 1: gather macro centers + boundary nodes into contiguous padded
// arrays in workspace. Pads [Nreal, Npad) get zero size/area and far-away,
// distinct positions -> zero contribution.
// ---------------------------------------------------------------------------
__global__ __launch_bounds__(256) void setup_kernel(
    const float* __restrict__ pos,
    const float* __restrict__ msx,
    const float* __restrict__ msy,
    const float* __restrict__ bpx,
    const float* __restrict__ bpy,
    const int*   __restrict__ midx,
    int m, int nreal, int npad, int num_nodes,
    float* __restrict__ X, float* __restrict__ Y,
    float* __restrict__ SX, float* __restrict__ SY,
    float* __restrict__ AR)
{
    int t = blockIdx.x * 256 + threadIdx.x;
    if (t >= npad) return;
    float x, y, sx, sy;
    if (t < m) {
        int id = midx[t];
        sx = msx[t];
        sy = msy[t];
        x  = pos[id]             + 0.5f * sx;
        y  = pos[num_nodes + id] + 0.5f * sy;
    } else if (t < nreal) {
        int b = t - m;
        sx = msx[t];
        sy = msy[t];
        x  = bpx[b];
        y  = bpy[b];
    } else {
        sx = 0.0f; sy = 0.0f;
        x  = 1.0e8f + (float)t * 1000.0f;   // distinct, far away
        y  = 1.0e8f + (float)t * 1000.0f;
    }
    X[t] = x; Y[t] = y; SX[t] = sx; SY[t] = sy; AR[t] = sx * sy;
}

// ---------------------------------------------------------------------------
// Kernel 2: full pairwise sum over a 16(i) x 256(j) tile per block.
// Wave-level reduction uses V_WMMA_F32_16X16X4_F32 (A = ones, B = partials)
// to fold acc[n] + acc[n+16] in one matrix op, then 4 shfl_xor steps.
// Per-block partial is written (no float atomics -> deterministic).
// ---------------------------------------------------------------------------
__global__ __launch_bounds__(256) void pair_kernel(
    const float* __restrict__ X, const float* __restrict__ Y,
    const float* __restrict__ SX, const float* __restrict__ SY,
    const float* __restrict__ AR,
    float* __restrict__ partials, int npad)
{
    __shared__ float six[16], siy[16], ssx[16], ssy[16], sar[16];
    __shared__ float wsum[8];

    const int tid = threadIdx.x;
    const int j   = blockIdx.x * 256 + tid;
    const int i0  = blockIdx.y * 16;

    if (tid < 16) {
        int i = i0 + tid;
        six[tid] = X[i];  siy[tid] = Y[i];
        ssx[tid] = SX[i]; ssy[tid] = SY[i];
        sar[tid] = AR[i];
    }
    __syncthreads();

    const float xj = X[j],  yj = Y[j];
    const float sxj = SX[j], syj = SY[j];
    const float aj = AR[j];

    float acc = 0.0f;
#pragma unroll
    for (int ii = 0; ii < 16; ++ii) {
        float dx = fabsf(six[ii] - xj);
        float dy = fabsf(siy[ii] - yj);
        float sx = 0.5f * (ssx[ii] + sxj);
        float sy = 0.5f * (ssy[ii] + syj);
        float px = bell_f(dx, sx);
        float py = bell_f(dy, sy);
        acc += (sar[ii] + aj) * px * py;
    }

    // ---- wave reduction via WMMA: D[m,n] = sum_k B[k,n]  (A = ones) ----
    // B VGPR0 holds the 32 lane partials across two K-rows, so every column n
    // of D equals acc[n] + acc[n+16]. EXEC is all-ones here (no divergence).
    v2f a; a.x = 1.0f; a.y = 1.0f;      // A = ones(16x4)
    v2f b; b.x = acc;  b.y = 0.0f;      // partials in B's first VGPR
    v8f c = {};
    c = __builtin_amdgcn_wmma_f32_16x16x4_f32(
            false, a, false, b, (short)0, c, false, false);
    float v = c[0];                      // lane L holds colsum_{L%16}
#pragma unroll
    for (int off = 1; off < 16; off <<= 1)
        v += __shfl_xor(v, off, 32);     // sum the 16 distinct column sums

    const int lane = tid & 31;
    const int wave = tid >> 5;
    if (lane == 0) wsum[wave] = v;
    __syncthreads();

    if (tid == 0) {
        float s = 0.0f;
#pragma unroll
        for (int w = 0; w < 8; ++w) s += wsum[w];   // fixed order: deterministic
        partials[blockIdx.y * gridDim.x + blockIdx.x] = s;
    }
}

// ---------------------------------------------------------------------------
// Kernel 3: deterministic final reduction + loss formula.
// triu sum = 0.5*S_full - A_tot  (diagonal pair (i,i) contributes 2*area_i).
// ---------------------------------------------------------------------------
__global__ __launch_bounds__(256) void finalize_kernel(
    const float* __restrict__ partials, int nblocks,
    const float* __restrict__ AR, int nreal,
    float* __restrict__ out)
{
    __shared__ float red[256];
    const int tid = threadIdx.x;

    float s = 0.0f;
    for (int k = tid; k < nblocks; k += 256) s += partials[k];
    red[tid] = s;
    __syncthreads();
    for (int w = 128; w > 0; w >>= 1) {
        if (tid < w) red[tid] += red[tid + w];
        __syncthreads();
    }
    float S_full = red[0];
    __syncthreads();

    float at = 0.0f;
    for (int k = tid; k < nreal; k += 256) at += AR[k];
    red[tid] = at;
    __syncthreads();
    for (int w = 128; w > 0; w >>= 1) {
        if (tid < w) red[tid] += red[tid + w];
        __syncthreads();
    }

    if (tid == 0) {
        float A_tot  = red[0];
        float S_triu = 0.5f * S_full - A_tot;
        float r      = S_triu / A_tot;
        out[0] = r * r;
    }
}

extern "C" void kernel_launch(void* const* d_in, const int* in_sizes, int n_in,
                              void* d_out, int out_size, void* d_ws, size_t ws_size,
                              hipStream_t stream) {
    const float* pos  = (const float*)d_in[0];
    const float* msx  = (const float*)d_in[1];
    const float* msy  = (const float*)d_in[2];
    const float* bpx  = (const float*)d_in[3];
    const float* bpy  = (const float*)d_in[4];
    const int*   midx = (const int*)d_in[5];

    const int num_nodes = in_sizes[0] / 2;
    const int nreal     = in_sizes[1];      // macros + boundary nodes (4252)
    const int m         = in_sizes[5];      // movable macros (4000)
    const int npad      = ((nreal + 255) / 256) * 256;   // 4352

    float* ws = (float*)d_ws;
    float* X  = ws;
    float* Y  = X  + npad;
    float* SX = Y  + npad;
    float* SY = SX + npad;
    float* AR = SY + npad;
    float* PART = AR + npad;

    const int jblocks = npad / 256;   // 17
    const int itiles  = npad / 16;    // 272
    const int nblocks = jblocks * itiles;

    setup_kernel<<<npad / 256, 256, 0, stream>>>(
        pos, msx, msy, bpx, bpy, midx, m, nreal, npad, num_nodes,
        X, Y, SX, SY, AR);

    pair_kernel<<<dim3(jblocks, itiles), 256, 0, stream>>>(
        X, Y, SX, SY, AR, PART, npad);

    finalize_kernel<<<1, 256, 0, stream>>>(
        PART, nblocks, AR, nreal, (float*)d_out);
}